// DDSP_70153995813593
// MI455X (gfx1250) — compile-verified
//
#include <hip/hip_runtime.h>
#include <math.h>

// ---------------------------------------------------------------------------
// DDSP synth + reverb for MI455X (gfx1250, wave32).
// Dominant cost: 8 x 51200 x 16000-tap causal reverb convolution = 13.1 GFLOP
// f32, all data L2-resident -> compute bound -> V_WMMA_F32_16X16X4_F32 as a
// Toeplitz GEMM. Impulse staged into LDS via the Tensor Data Mover (TDM).
// ---------------------------------------------------------------------------

#define Bn      8
#define Fn      400
#define NHn     100
#define NBn     65
#define BLK     128
#define LTOT    51200          // F * BLOCK samples per batch
#define IMPLEN  16000          // reverb impulse length (SR)
#define IMP_N   16128          // padded impulse buffer: [16 zeros][16000 taps][zeros]
#define SREV_N  67584          // per-batch reversed+zero-padded signal buffer
#define PI_F    3.14159265358979323846f

typedef __attribute__((ext_vector_type(8))) float        v8f;
typedef __attribute__((ext_vector_type(2))) float        v2f;
typedef __attribute__((ext_vector_type(4))) unsigned int v4u;
typedef __attribute__((ext_vector_type(4))) int          v4i;
typedef __attribute__((ext_vector_type(8))) int          v8i;

#if defined(__has_builtin)
#if __has_builtin(__builtin_amdgcn_tensor_load_to_lds) && __has_builtin(__builtin_amdgcn_s_wait_tensorcnt)
#define USE_TDM 1
#endif
#endif
#ifndef USE_TDM
#define USE_TDM 0
#endif

__device__ __forceinline__ float sigmoidf_(float x) { return 1.0f / (1.0f + expf(-x)); }
__device__ __forceinline__ float scalefn_(float x) {
    // 2 * sigmoid(x)^ln(10) + 1e-7
    return 2.0f * powf(sigmoidf_(x), 2.302585093f) + 1e-7f;
}

// ---------------------------------------------------------------------------
// Kernel A: per-frame prep. One block per (b,f), 128 threads.
//  - normalized harmonic amplitudes (anti-alias mask, sum-normalize, *total)
//  - 128-tap noise IR: irfft of real 65-bin spectrum + rolled-Hann window
//    ir[t] = (1/128)(X0 + 2*sum_{k=1..63} Xk cos(2pi k t/128) + X64 (-1)^t)
//            * (0.5 + 0.5 cos(2pi t/128))
//  - per-frame phase increment w = 2*pi*pitch/sr
// ---------------------------------------------------------------------------
__global__ __launch_bounds__(128) void frame_prep(
    const float* __restrict__ amp_param, const float* __restrict__ noise_param,
    const float* __restrict__ pitch, const int* __restrict__ srp,
    float* __restrict__ amps, float* __restrict__ irbuf, float* __restrict__ wbuf)
{
    __shared__ float s_param[104];
    __shared__ float s_np[72];
    __shared__ float s_red[128];

    const int bf = blockIdx.x;
    const int t  = threadIdx.x;
    const float srf = (float)(*srp);

    if (t < 101) s_param[t] = scalefn_(amp_param[bf * 101 + t]);
    if (t < 65)  s_np[t]    = scalefn_(noise_param[bf * 65 + t] - 5.0f);
    __syncthreads();

    const float pv = pitch[bf];
    float aval = 0.0f;
    if (t >= 1 && t <= 100) {
        const float aa = (pv * (float)t < 0.5f * srf) ? 1.0001f : 0.0001f;
        aval = s_param[t] * aa;
    }
    s_red[t] = aval;
    __syncthreads();
    for (int s = 64; s > 0; s >>= 1) {
        if (t < s) s_red[t] += s_red[t + s];
        __syncthreads();
    }
    if (t >= 1 && t <= 100)
        amps[bf * NHn + (t - 1)] = aval * (s_param[0] / s_red[0]);

    // irfft + window (t = tap index 0..127)
    const float w0 = 2.0f * PI_F / 128.0f;
    float acc = s_np[0];
    #pragma unroll 4
    for (int k = 1; k < 64; ++k)
        acc += 2.0f * s_np[k] * cosf(w0 * (float)(k * t));
    acc += s_np[64] * ((t & 1) ? -1.0f : 1.0f);
    irbuf[bf * BLK + t] = (acc * (1.0f / 128.0f)) * (0.5f + 0.5f * cosf(w0 * (float)t));

    if (t == 0) wbuf[bf] = 2.0f * PI_F * pv / srf;
}

// ---------------------------------------------------------------------------
// Kernel B: exclusive prefix scan of per-frame phase (400 frames per batch).
// omega[b, f*128+t] = sum_{f'<f} 128*w[f'] + (t+1)*w[f]
// ---------------------------------------------------------------------------
__global__ void scan_phase(const float* __restrict__ wbuf, float* __restrict__ basebuf)
{
    if (threadIdx.x != 0) return;
    const int b = blockIdx.x;
    float acc = 0.0f;
    for (int f = 0; f < Fn; ++f) {
        basebuf[b * Fn + f] = acc;
        acc += 128.0f * wbuf[b * Fn + f];
    }
}

// ---------------------------------------------------------------------------
// Kernel C: build zero-padded reverb impulse.
// impbuf[16+i]: i==0 -> 1.0 ; 1<=i<16000 -> rn[i]*exp(-softplus(-decay)*i/sr*500)*sigmoid(wet)
// ---------------------------------------------------------------------------
__global__ void impulse_build(const float* __restrict__ rn,
                              const float* __restrict__ decay,
                              const float* __restrict__ wet,
                              const int* __restrict__ srp,
                              float* __restrict__ impbuf)
{
    const int j = blockIdx.x * 256 + threadIdx.x;
    if (j >= IMP_N) return;
    const int i = j - 16;
    float v = 0.0f;
    if (i == 0) {
        v = 1.0f;
    } else if (i > 0 && i < IMPLEN) {
        const float sp  = logf(1.0f + expf(-decay[0]));   // softplus(-decay)
        const float srf = (float)(*srp);
        const float env = expf(-sp * ((float)i / srf) * 500.0f);
        v = rn[i] * env * sigmoidf_(wet[0]);
    }
    impbuf[j] = v;
}

__global__ void zero_buf(float* __restrict__ p, int n)
{
    const int j = blockIdx.x * 256 + threadIdx.x;
    if (j < n) p[j] = 0.0f;
}

// ---------------------------------------------------------------------------
// Kernel D: synthesis. One block per (b,f), thread t = sample in frame.
//  harmonic: 100-harmonic sine bank via Chebyshev recurrence
//  noise:    causal 128-tap conv of noise frame with IR (both LDS-staged)
// Result is written REVERSED into srev so the WMMA reverb kernel gets
// ascending per-lane addresses for its B fragments:
//   srev[b][LTOT - n] = sig[b][n],  zeros elsewhere.
// ---------------------------------------------------------------------------
__global__ __launch_bounds__(128) void synth(
    const float* __restrict__ noise, const float* __restrict__ amps,
    const float* __restrict__ irbuf, const float* __restrict__ wbuf,
    const float* __restrict__ basebuf, float* __restrict__ srev)
{
    __shared__ float s_noise[BLK];
    __shared__ float s_ir[BLK];
    __shared__ float s_amps[NHn];

    const int bf = blockIdx.x;
    const int t  = threadIdx.x;
    const int b  = bf / Fn;
    const int f  = bf - b * Fn;

    s_noise[t] = noise[bf * BLK + t];
    s_ir[t]    = irbuf[bf * BLK + t];
    if (t < NHn) s_amps[t] = amps[bf * NHn + t];
    __syncthreads();

    const float w     = wbuf[bf];
    const float omega = basebuf[bf] + (float)(t + 1) * w;

    // sin(h*omega), h = 1..100 by recurrence sin((h+1)w)=2cos(w)sin(hw)-sin((h-1)w)
    const float c2 = 2.0f * cosf(omega);
    float prev = 0.0f, cur = sinf(omega), harm = 0.0f;
    #pragma unroll 4
    for (int h = 0; h < NHn; ++h) {
        harm += s_amps[h] * cur;
        const float nxt = c2 * cur - prev;
        prev = cur; cur = nxt;
    }

    float nz = 0.0f;
    for (int m = 0; m <= t; ++m) nz += s_noise[m] * s_ir[t - m];

    const int n = f * BLK + t;
    srev[(size_t)b * SREV_N + (LTOT - n)] = harm + nz;
}

// ---------------------------------------------------------------------------
// Kernel E: reverb convolution as Toeplitz GEMM on the matrix core.
//   out[n0 + row + 16*col] = sum_p imp[row+p] * sig[n0 + 16*col - p]
//   A[row,kd] = impbuf[16 + row + pj + kd]   (LDS, ds_load)
//   B[kd,col] = srev[(LTOT - n0 - 16*col) + pj + kd]   (global, L2-resident)
// p chunked by 4 -> V_WMMA_F32_16X16X4_F32, 4008 K-steps per 256-output tile.
// 400 blocks x 4 waves; each wave owns one 16x16 tile (256 outputs).
// Impulse staged into LDS with one TENSOR_LOAD_TO_LDS DMA per block.
// ---------------------------------------------------------------------------
__global__ __launch_bounds__(128) void reverb_wmma(
    const float* __restrict__ impbuf, const float* __restrict__ srev,
    float* __restrict__ out)
{
    __shared__ float s_imp[IMP_N];

    const int tid = threadIdx.x;

#if USE_TDM
    // Tensor Data Mover: one DMA of the 16128-float (64.5 KB) padded impulse
    // into LDS, issued by wave 0 only (uniform branch skips other waves).
    // D# per ISA ch.8: group0 = {count=1, lds_addr, global_addr, type=2},
    // group1 = {data_size=4B, tensor_dim0=tile_dim0=16128, dim1=1, stride0=16128}.
    if (tid == 0) {
        const unsigned long long ga = (unsigned long long)(uintptr_t)impbuf;
        const unsigned int lds_addr = (unsigned int)(uintptr_t)&s_imp[0]; // LDS offset (low 32b of flat addr)
        v4u g0;
        g0.x = 1u;                                   // count=1, is_restore=0, gather off
        g0.y = lds_addr;                             // lds_addr [63:32]
        g0.z = (unsigned int)(ga & 0xFFFFFFFFu);     // global_addr [95:64]
        g0.w = (unsigned int)((ga >> 32) & 0x01FFFFFFu) | (2u << 30); // addr msbs + type=2
        v8i g1;
        g1[0] = 0x00020000;                          // workgroup_mask=0, data_size=2 (4B)
        g1[1] = (int)((IMP_N & 0xFFFF) << 16);       // tensor_dim0 low16 at [63:48]
        g1[2] = (int)(1u << 16);                     // tensor_dim0 hi=0, tensor_dim1=1
        g1[3] = (int)((IMP_N & 0xFFFF) << 16);       // tile_dim0 = 16128 at [127:112]
        g1[4] = 1;                                   // tile_dim1 = 1
        g1[5] = IMP_N;                               // tensor_dim0_stride low 32
        g1[6] = 0;
        g1[7] = 0;
        v4i gz = { 0, 0, 0, 0 };
#if __clang_major__ >= 23
        v8i gz8 = { 0, 0, 0, 0, 0, 0, 0, 0 };
        __builtin_amdgcn_tensor_load_to_lds(g0, g1, gz, gz, gz8, 0);
#else
        __builtin_amdgcn_tensor_load_to_lds(g0, g1, gz, gz, 0);
#endif
        __builtin_amdgcn_s_wait_tensorcnt(0);
    }
#else
    {
        const float4* src4 = (const float4*)impbuf;
        float4* dst4 = (float4*)s_imp;
        for (int j = tid; j < IMP_N / 4; j += 128) dst4[j] = src4[j];
    }
#endif
    __syncthreads();

    const int b    = blockIdx.x / 50;
    const int tg   = blockIdx.x % 50;
    const int wave = tid >> 5;
    const int lane = tid & 31;
    const int lo   = lane & 15;   // A: row (M) / B: col (N)
    const int hi   = lane >> 4;   // selects K pair {0,1} vs {2,3}
    const int n0   = (tg * 4 + wave) * 256;

    const float* srb  = srev + (size_t)b * SREV_N;
    const int aBase = 16 + lo + 2 * hi;               // + pj
    const int bBase = (LTOT - n0 - 16 * lo) + 2 * hi; // + pj

    v8f c = {};
    #pragma unroll 8
    for (int pj = -16; pj < 16016; pj += 4) {
        v2f a, bm;
        a.x  = s_imp[aBase + pj];         // A[row, K]   (VGPR0: K=0 / K=2)
        a.y  = s_imp[aBase + pj + 1];     //             (VGPR1: K=1 / K=3)
        bm.x = srb[bBase + pj];           // B[K, col]
        bm.y = srb[bBase + pj + 1];
        c = __builtin_amdgcn_wmma_f32_16x16x4_f32(
                /*neg_a=*/false, a, /*neg_b=*/false, bm,
                /*c_mod=*/(short)0, c, /*reuse_a=*/false, /*reuse_b=*/false);
    }

    // D layout: VGPR r -> M = r + 8*hi, N = lo  =>  n = n0 + 16*lo + 8*hi + r
    float* op = out + (size_t)b * LTOT + n0 + 16 * lo + 8 * hi;
    float4 r0 = { c[0], c[1], c[2], c[3] };
    float4 r1 = { c[4], c[5], c[6], c[7] };
    *(float4*)(op)     = r0;
    *(float4*)(op + 4) = r1;
}

// ---------------------------------------------------------------------------
// Host launcher
// ---------------------------------------------------------------------------
extern "C" void kernel_launch(void* const* d_in, const int* in_sizes, int n_in,
                              void* d_out, int out_size, void* d_ws, size_t ws_size,
                              hipStream_t stream)
{
    (void)in_sizes; (void)n_in; (void)out_size; (void)ws_size;

    const float* amp_param   = (const float*)d_in[0];
    const float* noise_param = (const float*)d_in[1];
    const float* pitch       = (const float*)d_in[2];
    const float* noise       = (const float*)d_in[3];
    const float* rnoise      = (const float*)d_in[4];
    const float* decay       = (const float*)d_in[5];
    const float* wet         = (const float*)d_in[6];
    const int*   srp         = (const int*)d_in[7];
    float* out = (float*)d_out;

    // Workspace layout (floats): 1,292,800 total (~4.9 MB)
    float* ws      = (float*)d_ws;
    float* amps    = ws;                         // 8*400*100 = 320000
    float* wbuf    = amps    + 320000;           // 3200
    float* basebuf = wbuf    + 3200;             // 3200
    float* irbuf   = basebuf + 3200;             // 8*400*128 = 409600
    float* impbuf  = irbuf   + 409600;           // 16128 (16-byte aligned)
    float* srev    = impbuf  + IMP_N;            // 8*67584 = 540672

    frame_prep<<<dim3(Bn * Fn), dim3(128), 0, stream>>>(
        amp_param, noise_param, pitch, srp, amps, irbuf, wbuf);
    scan_phase<<<dim3(Bn), dim3(32), 0, stream>>>(wbuf, basebuf);
    impulse_build<<<dim3((IMP_N + 255) / 256), dim3(256), 0, stream>>>(
        rnoise, decay, wet, srp, impbuf);
    zero_buf<<<dim3((Bn * SREV_N + 255) / 256), dim3(256), 0, stream>>>(
        srev, Bn * SREV_N);
    synth<<<dim3(Bn * Fn), dim3(128), 0, stream>>>(
        noise, amps, irbuf, wbuf, basebuf, srev);
    reverb_wmma<<<dim3(Bn * 50), dim3(128), 0, stream>>>(impbuf, srev, out);
}